// VecLocal2d_64450279243939
// MI455X (gfx1250) — compile-verified
//
#include <hip/hip_runtime.h>

// CDNA5 / gfx1250, wave32. fp32 locally-connected conv:
// per-location GEMM [320 x 288] x [288 x 64] via V_WMMA_F32_16X16X4_F32,
// weight staged to LDS with GLOBAL_LOAD_ASYNC_TO_LDS_B128 when available.

typedef __attribute__((ext_vector_type(2))) float v2f;
typedef __attribute__((ext_vector_type(4))) float v4f;
typedef __attribute__((ext_vector_type(8))) float v8f;
typedef int v4i_ __attribute__((vector_size(4 * sizeof(int))));

#define GLOBAL_AS __attribute__((address_space(1)))
#define LDS_AS    __attribute__((address_space(3)))

#define NBF   320      // B*CAT
#define CIN   32
#define HW    32
#define OCH   64
#define KTOT  288      // C*3*3
#define KC    72       // K chunk (multiple of 9 -> clean (c,p,q) cells)
#define LDST  76       // 72 + 4 pad floats (76 mod 64 = 12, gcd 4 -> bank spread)
#define MCH   64       // batch rows per block
#define OSTR  (OCH*HW*HW)   // 65536, out/bias batch stride

#if __has_builtin(__builtin_amdgcn_global_load_async_to_lds_b128)
#define USE_ASYNC 1
#else
#define USE_ASYNC 0
#endif

__device__ __forceinline__ void wait_async0() {
#if __has_builtin(__builtin_amdgcn_s_wait_asynccnt)
  __builtin_amdgcn_s_wait_asynccnt(0);
#else
  asm volatile("s_wait_asynccnt 0x0" ::: "memory");
#endif
}

__device__ __forceinline__ void store_acc(float* __restrict__ out,
                                          const float* __restrict__ bias,
                                          const v8f& acc, int bfrow0, int o, int s) {
  int cb = bfrow0 % 10;
  #pragma unroll
  for (int r = 0; r < 8; ++r) {
    int bf  = bfrow0 + r;
    int cat = cb + r;
    if (cat >= 10) cat -= 10;
    out[(size_t)bf * OSTR + o * (HW * HW) + s] =
        acc[r] + bias[cat * OSTR + o * (HW * HW) + s];
  }
}

__global__ __launch_bounds__(128)
void local2d_wmma_kernel(const float* __restrict__ x,
                         const float* __restrict__ wgt,
                         const float* __restrict__ bias,
                         float* __restrict__ out) {
  __shared__ float sA[MCH * LDST];   // im2col rows   [m][k]
  __shared__ float sB[OCH * LDST];   // weight rows   [o][k]

  const int s    = blockIdx.x;            // spatial location
  const int h    = s >> 5;
  const int w    = s & 31;
  const int mb   = blockIdx.y;            // batch chunk
  const int tid  = threadIdx.x;
  const int lane = tid & 31;
  const int wave = tid >> 5;              // 4 waves

  // 2x2 register tile per wave: 16 tiles = 4(M) x 4(N), 4 per wave
  const int obase = (wave & 1) * 32;      // ntiles: obase, obase+16
  const int mbase = (wave >> 1) * 32;     // mtiles: mbase, mbase+16
  const int mrow  = lane & 15;
  const int khalf = (lane >> 4) * 2;      // K offset per half-wave

  v8f acc00 = {}, acc01 = {}, acc10 = {}, acc11 = {};

  const float* wloc = wgt + (size_t)s * (OCH * KTOT);   // [64][288]
  const int bf0 = mb * MCH;

  // staging roles (128 threads): 2 threads per row, fixed tap pattern
  const int srow  = tid >> 1;             // 0..63
  const int shalf = tid & 1;              // half of the K range

  for (int ch = 0; ch < KTOT / KC; ++ch) {
    const int kc    = ch * KC;
    const int cbase = ch * 8;             // kc/9
    __syncthreads();

    // ---- stage B: contiguous [64][72] copy, async DMA to LDS ----
    {
      const float* grow = wloc + srow * KTOT + kc + shalf * 36;  // 9 float4
      float*       lrow = &sB[srow * LDST + shalf * 36];
      #pragma unroll
      for (int f4 = 0; f4 < 9; ++f4) {
#if USE_ASYNC
        __builtin_amdgcn_global_load_async_to_lds_b128(
            (GLOBAL_AS v4i_*)(grow + f4 * 4),
            (LDS_AS v4i_*)(lrow + f4 * 4),
            0, 0);
#else
        *(v4f*)(lrow + f4 * 4) = *(const v4f*)(grow + f4 * 4);
#endif
      }
    }

    // ---- stage A: im2col gather, no divisions (KC multiple of 9) ----
    {
      const float* xb = x + (size_t)(bf0 + srow) * (CIN * HW * HW);
      float*       la = &sA[srow * LDST + shalf * 36];
      #pragma unroll
      for (int cs = 0; cs < 4; ++cs) {
        const int c = cbase + shalf * 4 + cs;
        const float* xc = xb + c * (HW * HW);
        #pragma unroll
        for (int p = 0; p < 3; ++p) {
          const int y = h + p - 1;
          const bool yok = (unsigned)y < HW;
          const float* xr = xc + y * HW + (w - 1);
          #pragma unroll
          for (int q = 0; q < 3; ++q) {
            const int xw = w + q - 1;
            float v = 0.0f;
            if (yok && (unsigned)xw < HW) v = xr[q];
            la[cs * 9 + p * 3 + q] = v;
          }
        }
      }
    }

#if USE_ASYNC
    wait_async0();
#endif
    __syncthreads();

    // ---- WMMA main loop: 1.0 LDS b64 load per WMMA ----
    #pragma unroll 4
    for (int kk4 = 0; kk4 < KC; kk4 += 4) {
      const int kf = kk4 + khalf;   // even -> 8B-aligned LDS b64
      v2f b0 = *(const v2f*)&sB[(obase      + mrow) * LDST + kf];
      v2f b1 = *(const v2f*)&sB[(obase + 16 + mrow) * LDST + kf];
      v2f a0 = *(const v2f*)&sA[(mbase      + mrow) * LDST + kf];
      v2f a1 = *(const v2f*)&sA[(mbase + 16 + mrow) * LDST + kf];
      acc00 = __builtin_amdgcn_wmma_f32_16x16x4_f32(false, a0, false, b0,
                                                    (short)0, acc00, false, false);
      acc01 = __builtin_amdgcn_wmma_f32_16x16x4_f32(false, a0, false, b1,
                                                    (short)0, acc01, false, false);
      acc10 = __builtin_amdgcn_wmma_f32_16x16x4_f32(false, a1, false, b0,
                                                    (short)0, acc10, false, false);
      acc11 = __builtin_amdgcn_wmma_f32_16x16x4_f32(false, a1, false, b1,
                                                    (short)0, acc11, false, false);
    }
  }

  // ---- epilogue: VGPR r -> M = r + 8*(lane/16), N = lane%16 ----
  const int o0 = obase + mrow;
  const int o1 = o0 + 16;
  const int mhalf = (lane >> 4) * 8;
  const int row0 = bf0 + mbase + mhalf;
  store_acc(out, bias, acc00, row0,      o0, s);
  store_acc(out, bias, acc01, row0,      o1, s);
  store_acc(out, bias, acc10, row0 + 16, o0, s);
  store_acc(out, bias, acc11, row0 + 16, o1, s);
}

extern "C" void kernel_launch(void* const* d_in, const int* in_sizes, int n_in,
                              void* d_out, int out_size, void* d_ws, size_t ws_size,
                              hipStream_t stream) {
  (void)in_sizes; (void)n_in; (void)out_size; (void)d_ws; (void)ws_size;
  const float* x    = (const float*)d_in[0];   // [32,10,32,32,32]
  const float* wgt  = (const float*)d_in[1];   // [32,32,64,32,3,3]
  const float* bias = (const float*)d_in[2];   // [10,64,32,32]
  float* out = (float*)d_out;                  // [32,10,64,32,32]

  dim3 grid(HW * HW, NBF / MCH);               // (1024 locations, 5 batch chunks)
  dim3 block(128);                             // 4 waves (wave32)
  local2d_wmma_kernel<<<grid, block, 0, stream>>>(x, wgt, bias, out);
}